// TripletMiner_16131897164379
// MI455X (gfx1250) — compile-verified
//
#include <hip/hip_runtime.h>
#include <stdint.h>

// Problem constants (from reference: N=16384, D=128, C=1000)
#define NA 16384      // number of anchors / rows
#define DIM 128       // embedding dim
#define F4_PER_ROW (DIM / 4)          // 32 float4 per row -> one wave32 per row
#define SLICE (NA * DIM)              // floats per triplet slice

// ---------------------------------------------------------------------------
// Kernel 1: mining. Each block stages ALL 16384 labels (64 KB) into LDS via
// CDNA5 async global->LDS B128 copies (ASYNCcnt), then each thread scans
// forward from its anchor for the first same-label and first diff-label index.
// Emits packed u16 pos/neg indices (N <= 16384 fits u16) and the float valid
// mask directly into its final output slot.
// ---------------------------------------------------------------------------
__global__ __launch_bounds__(256) void TripletMiner_mine_kernel(
    const int* __restrict__ labels,
    unsigned short* __restrict__ pos16,
    unsigned short* __restrict__ neg16,
    float* __restrict__ validf)
{
    __shared__ int lds_labels[NA];   // 64 KB (LDS is 320 KB/WGP on CDNA5)

    const int tid = threadIdx.x;

    // ---- async stage: 256 threads * 16 iters * 16 B = 64 KB ----
    {
        unsigned long long gbase = (unsigned long long)labels;
        #pragma unroll
        for (int k = 0; k < 16; ++k) {
            int byteoff = (k * 256 + tid) * 16;
            // Flat->LDS mapping uses addr[31:0]; truncating the generic
            // pointer of a __shared__ object yields the LDS byte address.
            unsigned lds_addr =
                (unsigned)(size_t)((char*)lds_labels + byteoff);
            asm volatile(
                "global_load_async_to_lds_b128 %0, %1, %2"
                :
                : "v"(lds_addr), "v"(byteoff), "s"(gbase)
                : "memory");
        }
        asm volatile("s_wait_asynccnt 0x0" ::: "memory");
    }
    __syncthreads();

    const int i = blockIdx.x * 256 + tid;   // grid = 64 blocks covers NA
    const int myLab = lds_labels[i];

    int pos = -1, neg = -1;
    for (int j = i + 1; j < NA; ++j) {
        const int lj = lds_labels[j];
        if (lj == myLab) {
            if (pos < 0) { pos = j; if (neg >= 0) break; }
        } else {
            if (neg < 0) { neg = j; if (pos >= 0) break; }
        }
    }

    const bool v = (pos >= 0) && (neg >= 0);
    // argmax-of-all-false semantics: index 0 when absent (masked out anyway)
    pos16[i] = (unsigned short)(pos >= 0 ? pos : 0);
    neg16[i] = (unsigned short)(neg >= 0 ? neg : 0);
    validf[i] = v ? 1.0f : 0.0f;
}

// ---------------------------------------------------------------------------
// Kernel 2: gather + write. One wave32 per anchor row: lane L owns floats
// [4L..4L+3] => single global_load_b128 / global_store_b128 per matrix row.
// Anchor row is streaming; pos/neg rows are gathers that hit the 192 MB L2
// (embeddings = 8 MB). Everything scaled by the valid mask.
// ---------------------------------------------------------------------------
__global__ __launch_bounds__(256) void TripletMiner_gather_kernel(
    const float* __restrict__ emb,
    const unsigned short* __restrict__ pos16,
    const unsigned short* __restrict__ neg16,
    const float* __restrict__ validf,
    float* __restrict__ out)
{
    const int lane = threadIdx.x & 31;
    const int wave = threadIdx.x >> 5;
    const int i = blockIdx.x * 8 + wave;     // 8 waves per block, one anchor each

    const float v = validf[i];
    const int p = (int)pos16[i];
    const int n = (int)neg16[i];

    const float4* __restrict__ embv = (const float4*)emb;
    float4 a  = embv[i * F4_PER_ROW + lane];
    float4 pp = embv[p * F4_PER_ROW + lane];
    float4 nn = embv[n * F4_PER_ROW + lane];

    a.x *= v;  a.y *= v;  a.z *= v;  a.w *= v;
    pp.x *= v; pp.y *= v; pp.z *= v; pp.w *= v;
    nn.x *= v; nn.y *= v; nn.z *= v; nn.w *= v;

    float4* outv = (float4*)out;
    const int base = i * F4_PER_ROW + lane;
    const int slice4 = SLICE / 4;            // float4 per slice
    outv[0 * slice4 + base] = a;
    outv[1 * slice4 + base] = pp;
    outv[2 * slice4 + base] = nn;
}

// ---------------------------------------------------------------------------
// Launcher. d_in[0] = embeddings (float32, N*D), d_in[1] = labels (int).
// d_out = [3*N*D triplets][N valid] as float32. d_ws holds packed u16 indices.
// ---------------------------------------------------------------------------
extern "C" void kernel_launch(void* const* d_in, const int* in_sizes, int n_in,
                              void* d_out, int out_size, void* d_ws, size_t ws_size,
                              hipStream_t stream) {
    (void)in_sizes; (void)n_in; (void)out_size; (void)ws_size;

    const float* emb   = (const float*)d_in[0];
    const int* labels  = (const int*)d_in[1];
    float* out         = (float*)d_out;

    unsigned short* pos16 = (unsigned short*)d_ws;          // N * 2 bytes
    unsigned short* neg16 = pos16 + NA;                     // N * 2 bytes
    float* validf = out + 3 * SLICE;                        // final output slot

    TripletMiner_mine_kernel<<<NA / 256, 256, 0, stream>>>(
        labels, pos16, neg16, validf);

    TripletMiner_gather_kernel<<<NA / 8, 256, 0, stream>>>(
        emb, pos16, neg16, validf, out);
}